// KoopmanRotBlocks_50422916055375
// MI455X (gfx1250) — compile-verified
//
#include <hip/hip_runtime.h>
#include <cstdint>

#define INV_2PI 0.15915494309189535f   // v_sin/v_cos take revolutions (x -> sin(2*pi*x))
#define LOG2E   1.4426950408889634f    // v_exp_f32 is exp2

// B=32, T=256, D=128 (64 rotation pairs). Grid: 32 t-chunks per batch,
// each block = 8 t-rows x 32 pair-groups (one float4 = 2 pairs per lane).
__global__ __launch_bounds__(256) void koopman_rot_kernel(
    const float* __restrict__ z0,     // [32,128] interleaved (e,o) pairs
    const float* __restrict__ dt,     // [32,256]
    const float* __restrict__ logw,   // [64]
    float* __restrict__ out)          // [32,256,128]
{
    constexpr int T = 256;
    constexpr int D = 128;

    __shared__ float4 s_z[32];   // z0 row: 512 B ({ze0,zo0,ze1,zo1} per group)
    __shared__ float2 s_w[32];   // log_omega: 256 B

    const int b  = blockIdx.x >> 5;       // batch
    const int tc = blockIdx.x & 31;       // t-chunk
    const int tl = threadIdx.x >> 5;      // 0..7 local row (one wave per row)
    const int g  = threadIdx.x & 31;      // pair-group 0..31
    const int t  = tc * 8 + tl;

    // ---- CDNA5 async global->LDS staging (ASYNCcnt path), wave 0 only ----
    if (threadIdx.x < 32) {
        unsigned lds_z = (unsigned)(uintptr_t)(&s_z[g]);   // low 32 bits = LDS byte offset
        unsigned lds_w = (unsigned)(uintptr_t)(&s_w[g]);
        unsigned long long gz = (unsigned long long)(uintptr_t)(z0 + b * D + g * 4);
        unsigned long long gw = (unsigned long long)(uintptr_t)(logw + g * 2);
        asm volatile("global_load_async_to_lds_b128 %0, %1, off"
                     :: "v"(lds_z), "v"(gz) : "memory");
        asm volatile("global_load_async_to_lds_b64 %0, %1, off"
                     :: "v"(lds_w), "v"(gw) : "memory");
        asm volatile("s_wait_asynccnt 0x0" ::: "memory");
    }
    __syncthreads();

    const float dtv = dt[b * T + t];     // uniform across the wave

    const float4 z  = s_z[g];            // ds_load_b128
    const float2 lw = s_w[g];            // ds_load_b64

    // omega/(2pi) = exp2(log_omega * log2e) * (1/2pi); fold normalization once.
    const float w0  = __builtin_amdgcn_exp2f(lw.x * LOG2E) * INV_2PI;
    const float w1  = __builtin_amdgcn_exp2f(lw.y * LOG2E) * INV_2PI;
    const float tn0 = dtv * w0;
    const float tn1 = dtv * w1;
    const float s0  = __builtin_amdgcn_sinf(tn0);   // v_sin_f32 (TRANS, co-executes)
    const float c0  = __builtin_amdgcn_cosf(tn0);
    const float s1  = __builtin_amdgcn_sinf(tn1);
    const float c1  = __builtin_amdgcn_cosf(tn1);

    float4 r;
    r.x = c0 * z.x - s0 * z.y;   // zt_e0
    r.y = s0 * z.x + c0 * z.y;   // zt_o0
    r.z = c1 * z.z - s1 * z.w;   // zt_e1
    r.w = s1 * z.z + c1 * z.w;   // zt_o1

    float4* o = reinterpret_cast<float4*>(out + (size_t)(b * T + t) * D) + g;
    *o = r;                      // global_store_b128: 512 B per wave-row
}

extern "C" void kernel_launch(void* const* d_in, const int* in_sizes, int n_in,
                              void* d_out, int out_size, void* d_ws, size_t ws_size,
                              hipStream_t stream) {
    (void)in_sizes; (void)n_in; (void)out_size; (void)d_ws; (void)ws_size;
    const float* z0   = (const float*)d_in[0];   // (32,128)
    const float* dt   = (const float*)d_in[1];   // (32,256)
    const float* logw = (const float*)d_in[2];   // (64,)
    float* out = (float*)d_out;                  // (32,256,128)

    // 32 batches * 32 t-chunks = 1024 blocks, 256 threads (8 waves) each.
    koopman_rot_kernel<<<dim3(1024), dim3(256), 0, stream>>>(z0, dt, logw, out);
}